// MYGIN_67662914781224
// MI455X (gfx1250) — compile-verified
//
#include <hip/hip_runtime.h>

#define NN 100000
#define NE 1280000
#define D  64
#define BN_EPS 1e-5f

typedef __attribute__((ext_vector_type(2))) float v2f;
typedef __attribute__((ext_vector_type(8))) float v8f;

// ---------------------------------------------------------------------------
// Fold BatchNorm params into per-feature scale/shift:  y = s*x + t
// coef layout: [s0|t0|s1|t1|sb|tb], 64 floats each.
// ---------------------------------------------------------------------------
__global__ void gin_bn_coeffs(const float* g0, const float* be0, const float* m0, const float* v0,
                              const float* g1, const float* be1, const float* m1, const float* v1,
                              const float* gb, const float* bb, const float* mb, const float* vb,
                              float* coef) {
    int i = threadIdx.x;
    if (i < D) {
        float s, t;
        s = g0[i] * rsqrtf(v0[i] + BN_EPS); t = be0[i] - m0[i] * s;
        coef[0 * D + i] = s; coef[1 * D + i] = t;
        s = g1[i] * rsqrtf(v1[i] + BN_EPS); t = be1[i] - m1[i] * s;
        coef[2 * D + i] = s; coef[3 * D + i] = t;
        s = gb[i] * rsqrtf(vb[i] + BN_EPS); t = bb[i] - mb[i] * s;
        coef[4 * D + i] = s; coef[5 * D + i] = t;
    }
}

// ---------------------------------------------------------------------------
// Edge scatter: agg[dst] += pre(h[src]).  pre = relu(cs*x+ct) when apply!=0.
// One thread per (edge, 4-feature chunk); f32 HW atomics land in L2
// (agg is 25.6MB, fully L2-resident on MI455X's 192MB L2).
// ---------------------------------------------------------------------------
__global__ __launch_bounds__(256) void gin_scatter(
        const float* __restrict__ h, const int* __restrict__ src,
        const int* __restrict__ dst, float* __restrict__ agg,
        const float* __restrict__ cs, const float* __restrict__ ct, int apply) {
    long idx = (long)blockIdx.x * blockDim.x + threadIdx.x;
    const long total = (long)NE * 16;
    if (idx >= total) return;
    int e  = (int)(idx >> 4);
    int c4 = (int)(idx & 15);
    int s  = src[e];
    int d  = dst[e];
    float4 v = *(const float4*)(h + (long)s * D + c4 * 4);
    if (apply) {
        float4 a = *(const float4*)(cs + c4 * 4);
        float4 b = *(const float4*)(ct + c4 * 4);
        v.x = fmaxf(fmaf(a.x, v.x, b.x), 0.f);
        v.y = fmaxf(fmaf(a.y, v.y, b.y), 0.f);
        v.z = fmaxf(fmaf(a.z, v.z, b.z), 0.f);
        v.w = fmaxf(fmaf(a.w, v.w, b.w), 0.f);
    }
    float* ap = agg + (long)d * D + c4 * 4;
    unsafeAtomicAdd(ap + 0, v.x);
    unsafeAtomicAdd(ap + 1, v.y);
    unsafeAtomicAdd(ap + 2, v.z);
    unsafeAtomicAdd(ap + 3, v.w);
}

// ---------------------------------------------------------------------------
// Fused GIN MLP:  out = relu(bn(  (pre(h)+agg) @ Wa + ba )) @ Wb + bb
// pre = relu(ps*x+pt) when apply_pre!=0 (the inter-layer bn1+relu).
// Block = 128 threads = 4 waves; each wave owns a 16-row tile (64 rows/block).
// FP32 WMMA 16x16x4: A/B are v2f per lane (VGPR0 = K even, VGPR1 = K odd;
// lanes 16..31 carry K+2), C/D are v8f (row M = j + 8*(lane>>4), col N = lane&15).
// ---------------------------------------------------------------------------
#define APAD 66  // padded LDS row stride (floats): bank-conflict-free strided reads

__global__ __launch_bounds__(128) void gin_mlp_wmma(
        const float* __restrict__ hin, const float* __restrict__ agg,
        const float* __restrict__ Wa,  const float* __restrict__ ba,
        const float* __restrict__ sa,  const float* __restrict__ ta,
        const float* __restrict__ Wb,  const float* __restrict__ bbv,
        const float* __restrict__ ps,  const float* __restrict__ pt,
        int apply_pre, float* __restrict__ out) {
    // packed weights: sW[k/2][n] = {W[k][n], W[k+1][n]}  (matches B operand layout)
    __shared__ v2f sWa[32 * D];
    __shared__ v2f sWb[32 * D];
    __shared__ __align__(16) float sA[4][16][APAD];

    const int t    = threadIdx.x;
    const int base = blockIdx.x * 64;

    // ---- stage packed weights (once per block) ----
    for (int i = 0; i < 16; ++i) {
        int p  = t + i * 128;          // pair index 0..2047
        int k2 = p >> 6, n = p & 63;
        v2f wa, wb;
        wa[0] = Wa[(2 * k2) * D + n];  wa[1] = Wa[(2 * k2 + 1) * D + n];
        wb[0] = Wb[(2 * k2) * D + n];  wb[1] = Wb[(2 * k2 + 1) * D + n];
        sWa[p] = wa;
        sWb[p] = wb;
    }

    // ---- stage h2 = pre(h) + agg tile (64 rows x 64 cols), coalesced float4 ----
    for (int i = 0; i < 8; ++i) {
        int f4 = t + i * 128;          // float4 index 0..1023
        int r  = f4 >> 4;              // tile row 0..63
        int c4 = f4 & 15;
        long row = base + r;
        if (row >= NN) row = NN - 1;   // clamp; stores are masked later
        float4 hv = *(const float4*)(hin + row * D + c4 * 4);
        if (apply_pre) {
            float4 a = *(const float4*)(ps + c4 * 4);
            float4 b = *(const float4*)(pt + c4 * 4);
            hv.x = fmaxf(fmaf(a.x, hv.x, b.x), 0.f);
            hv.y = fmaxf(fmaf(a.y, hv.y, b.y), 0.f);
            hv.z = fmaxf(fmaf(a.z, hv.z, b.z), 0.f);
            hv.w = fmaxf(fmaf(a.w, hv.w, b.w), 0.f);
        }
        float4 av = *(const float4*)(agg + row * D + c4 * 4);
        float* dp = &sA[r >> 4][r & 15][c4 * 4];
        dp[0] = hv.x + av.x;
        dp[1] = hv.y + av.y;
        dp[2] = hv.z + av.z;
        dp[3] = hv.w + av.w;
    }
    __syncthreads();

    const int w    = t >> 5;
    const int lane = t & 31;
    const int sel  = lane >> 4;        // which K-pair half this lane carries
    const int lr   = lane & 15;

    // ---- GEMM1: u = h2 @ Wa + ba ----
    v8f a0, a1, a2, a3;
    {
        float b0 = ba[lr], b1 = ba[16 + lr], b2 = ba[32 + lr], b3 = ba[48 + lr];
        for (int j = 0; j < 8; ++j) { a0[j] = b0; a1[j] = b1; a2[j] = b2; a3[j] = b3; }
    }
    for (int ks = 0; ks < 16; ++ks) {
        v2f av = *(const v2f*)&sA[w][lr][4 * ks + 2 * sel];
        int kr = (2 * ks + sel) * D;
        a0 = __builtin_amdgcn_wmma_f32_16x16x4_f32(false, av, false, sWa[kr + lr],      (short)0, a0, false, false);
        a1 = __builtin_amdgcn_wmma_f32_16x16x4_f32(false, av, false, sWa[kr + 16 + lr], (short)0, a1, false, false);
        a2 = __builtin_amdgcn_wmma_f32_16x16x4_f32(false, av, false, sWa[kr + 32 + lr], (short)0, a2, false, false);
        a3 = __builtin_amdgcn_wmma_f32_16x16x4_f32(false, av, false, sWa[kr + 48 + lr], (short)0, a3, false, false);
    }

    // ---- fused BN + ReLU on accumulators (per output column) ----
    {
        float s0 = sa[lr],      t0 = ta[lr];
        float s1 = sa[16 + lr], t1 = ta[16 + lr];
        float s2 = sa[32 + lr], t2 = ta[32 + lr];
        float s3 = sa[48 + lr], t3 = ta[48 + lr];
        for (int j = 0; j < 8; ++j) {
            a0[j] = fmaxf(fmaf(s0, a0[j], t0), 0.f);
            a1[j] = fmaxf(fmaf(s1, a1[j], t1), 0.f);
            a2[j] = fmaxf(fmaf(s2, a2[j], t2), 0.f);
            a3[j] = fmaxf(fmaf(s3, a3[j], t3), 0.f);
        }
    }

    // ---- transpose D-layout -> A-layout via this wave's private LDS region ----
    for (int j = 0; j < 8; ++j) {
        int m = j + 8 * sel;
        sA[w][m][lr]      = a0[j];
        sA[w][m][16 + lr] = a1[j];
        sA[w][m][32 + lr] = a2[j];
        sA[w][m][48 + lr] = a3[j];
    }
    // single-wave region, DS ops are in-order within a wave: no barrier needed

    // ---- GEMM2: out = u @ Wb + bb ----
    v8f o0, o1, o2, o3;
    {
        float b0 = bbv[lr], b1 = bbv[16 + lr], b2 = bbv[32 + lr], b3 = bbv[48 + lr];
        for (int j = 0; j < 8; ++j) { o0[j] = b0; o1[j] = b1; o2[j] = b2; o3[j] = b3; }
    }
    for (int ks = 0; ks < 16; ++ks) {
        v2f av = *(const v2f*)&sA[w][lr][4 * ks + 2 * sel];
        int kr = (2 * ks + sel) * D;
        o0 = __builtin_amdgcn_wmma_f32_16x16x4_f32(false, av, false, sWb[kr + lr],      (short)0, o0, false, false);
        o1 = __builtin_amdgcn_wmma_f32_16x16x4_f32(false, av, false, sWb[kr + 16 + lr], (short)0, o1, false, false);
        o2 = __builtin_amdgcn_wmma_f32_16x16x4_f32(false, av, false, sWb[kr + 32 + lr], (short)0, o2, false, false);
        o3 = __builtin_amdgcn_wmma_f32_16x16x4_f32(false, av, false, sWb[kr + 48 + lr], (short)0, o3, false, false);
    }

    // ---- masked stores (lanes coalesce along the feature dimension) ----
    for (int j = 0; j < 8; ++j) {
        int  m   = j + 8 * sel;
        long row = base + w * 16 + m;
        if (row < NN) {
            float* op = out + row * D;
            op[lr]      = o0[j];
            op[16 + lr] = o1[j];
            op[32 + lr] = o2[j];
            op[48 + lr] = o3[j];
        }
    }
}

// ---------------------------------------------------------------------------
extern "C" void kernel_launch(void* const* d_in, const int* in_sizes, int n_in,
                              void* d_out, int out_size, void* d_ws, size_t ws_size,
                              hipStream_t stream) {
    const float* x   = (const float*)d_in[0];
    const int*   ei  = (const int*)d_in[1];
    const int*   src = ei;
    const int*   dst = ei + NE;
    const float* W0a = (const float*)d_in[2];
    const float* b0a = (const float*)d_in[3];
    const float* g0  = (const float*)d_in[4];
    const float* be0 = (const float*)d_in[5];
    const float* m0  = (const float*)d_in[6];
    const float* v0  = (const float*)d_in[7];
    const float* W0b = (const float*)d_in[8];
    const float* b0b = (const float*)d_in[9];
    const float* W1a = (const float*)d_in[10];
    const float* b1a = (const float*)d_in[11];
    const float* g1  = (const float*)d_in[12];
    const float* be1 = (const float*)d_in[13];
    const float* m1  = (const float*)d_in[14];
    const float* v1  = (const float*)d_in[15];
    const float* W1b = (const float*)d_in[16];
    const float* b1b = (const float*)d_in[17];
    const float* gb  = (const float*)d_in[18];
    const float* bbp = (const float*)d_in[19];
    const float* mb  = (const float*)d_in[20];
    const float* vb  = (const float*)d_in[21];

    float* wsf  = (float*)d_ws;
    float* agg  = wsf;                               // N*D floats
    float* coef = wsf + (size_t)NN * D;              // 6*64 floats
    float* s0 = coef + 0 * D, *t0 = coef + 1 * D;
    float* s1 = coef + 2 * D, *t1 = coef + 3 * D;
    float* sb = coef + 4 * D, *tb = coef + 5 * D;

    float* outp = (float*)d_out;                     // [N,D] layer-1 output
    float* z    = outp + (size_t)NN * D;             // [N,D] layer-0 output

    gin_bn_coeffs<<<1, 64, 0, stream>>>(g0, be0, m0, v0, g1, be1, m1, v1,
                                        gb, bbp, mb, vb, coef);

    const int sblocks = (int)(((long)NE * 16 + 255) / 256);
    const int mblocks = (NN + 63) / 64;

    // ---- layer 0 ----
    hipMemsetAsync(agg, 0, (size_t)NN * D * sizeof(float), stream);
    gin_scatter<<<sblocks, 256, 0, stream>>>(x, src, dst, agg, nullptr, nullptr, 0);
    gin_mlp_wmma<<<mblocks, 128, 0, stream>>>(x, agg, W0a, b0a, s0, t0, W0b, b0b,
                                              nullptr, nullptr, 0, z);

    // ---- layer 1 (bn1+relu folded into the gather/self paths) ----
    hipMemsetAsync(agg, 0, (size_t)NN * D * sizeof(float), stream);
    gin_scatter<<<sblocks, 256, 0, stream>>>(z, src, dst, agg, sb, tb, 1);
    gin_mlp_wmma<<<mblocks, 128, 0, stream>>>(z, agg, W1a, b1a, s1, t1, W1b, b1b,
                                              sb, tb, 1, outp);
}